// AttnDecoderRNN_72335839199421
// MI455X (gfx1250) — compile-verified
//
#include <hip/hip_runtime.h>
#include <hip/hip_bf16.h>
#include <math.h>

// ---------------------------------------------------------------------------
// AttnDecoderRNN single step on MI455X (gfx1250, wave32).
// Batch-1 GEMVs -> HBM-bandwidth bound (~560MB => ~24us at 23.3 TB/s).
// GEMVs run on V_WMMA_F32_16X16X4_F32 (full f32, deterministic RNE):
//   - one 16-row group per wave
//   - A streamed with non-temporal global_load_b128 (weights are use-once,
//     out_w alone is 412MB > 192MB L2)
//   - B (the x vector) staged once per block in LDS, broadcast ds reads
//   - 4 independent accumulators to pipeline the WMMAs
// ---------------------------------------------------------------------------

#define H_DIM 2048
#define S_DIM 2048
#define V_DIM 50257

typedef __attribute__((ext_vector_type(2))) float v2f;
typedef __attribute__((ext_vector_type(4))) float v4f;
typedef __attribute__((ext_vector_type(8))) float v8f;

// ---------------------------------------------------------------------------
// Kernel 1: partial column sums of encoder_outputs (S x H), 32 S-chunks,
// v4f loads. partials[sb*H + 4*c4 ...] = sum over 64 rows of enc[s*H + col].
// ---------------------------------------------------------------------------
__global__ void colsum_partial_kernel(const float* __restrict__ enc,
                                      float* __restrict__ partials) {
    const int c4 = blockIdx.x * blockDim.x + threadIdx.x;  // column/4 index
    const int sb = blockIdx.y;                             // S-chunk (32 chunks)
    const int rows = S_DIM / 32;                           // 64 rows per chunk
    const int s0 = sb * rows;
    const v4f* e = (const v4f*)enc;
    v4f acc = {0.f, 0.f, 0.f, 0.f};
    for (int s = 0; s < rows; ++s) {
        acc += e[(size_t)(s0 + s) * (H_DIM / 4) + c4];
    }
    ((v4f*)partials)[sb * (H_DIM / 4) + c4] = acc;
}

// ---------------------------------------------------------------------------
// Kernel 2: combine partials -> c2h[H..2H); gather embedding row -> c2h[0..H);
// write attn_weights output (all ones; S == H so one thread covers both).
// ---------------------------------------------------------------------------
__global__ void prep_combine_kernel(const float* __restrict__ emb,
                                    const int* __restrict__ tok,
                                    const float* __restrict__ partials,
                                    float* __restrict__ c2h,
                                    float* __restrict__ out_attnw) {
    const int c = blockIdx.x * blockDim.x + threadIdx.x;
    if (c < H_DIM) {
        float acc = 0.0f;
        for (int j = 0; j < 32; ++j) acc += partials[j * H_DIM + c];
        c2h[H_DIM + c] = acc;                         // attn_applied
        c2h[c] = emb[(size_t)tok[0] * H_DIM + c];     // embedded row
    }
    if (c < S_DIM) out_attnw[c] = 1.0f;               // softmax(singleton) == 1
}

// ---------------------------------------------------------------------------
// WMMA GEMV: y[m] = op(sum_k W[m,k]*x[k] + b[m]).
// One wave owns 16 rows. K-columns are freely re-assignable to WMMA slots as
// long as A and B agree:
//   per 16 columns: lane(row,hi) loads W row float4s at kb+4*hi and kb+8+4*hi;
//   .lo/.hi halves feed 4 WMMA steps; B comes from LDS with identical indexing.
// A-slot layout (ISA): lanes 0-15 = K slots {0,1}, lanes 16-31 = {2,3};
// C/D: VGPR j, lane 0 -> D[j][0] (rows 0..7), lane 16 -> D[8+j][0] (rows 8..15).
// ---------------------------------------------------------------------------
__global__ void gemv_wmma_kernel(const float* __restrict__ W,
                                 const float* __restrict__ x,
                                 const float* __restrict__ bias,
                                 float* __restrict__ y,
                                 int M, int K, int relu_flag) {
    extern __shared__ v4f sx4[];  // K floats staged once per block

    for (int i = threadIdx.x; i < (K >> 2); i += blockDim.x) {
        sx4[i] = ((const v4f*)x)[i];
    }
    __syncthreads();

    const int lane  = threadIdx.x & 31;
    const int wid   = threadIdx.x >> 5;
    const int wpb   = blockDim.x >> 5;
    const int gwarp = blockIdx.x * wpb + wid;
    const int nwarp = gridDim.x * wpb;
    const int row   = lane & 15;
    const int hi    = lane >> 4;
    const int groups = (M + 15) >> 4;

    for (int g = gwarp; g < groups; g += nwarp) {
        const int rbase = g << 4;
        int rr = rbase + row;
        if (rr > M - 1) rr = M - 1;  // clamp; stores guarded below

        const v4f* wrow4 = (const v4f*)(W + (size_t)rr * K);

        v8f acc0 = {0.f, 0.f, 0.f, 0.f, 0.f, 0.f, 0.f, 0.f};
        v8f acc1 = acc0, acc2 = acc0, acc3 = acc0;

        for (int kb = 0; kb < K; kb += 16) {       // K % 16 == 0 in all uses
            const int q = (kb >> 2) + hi;
            v4f a01 = __builtin_nontemporal_load(wrow4 + q);      // cols kb+4hi..+3
            v4f a23 = __builtin_nontemporal_load(wrow4 + q + 2);  // cols kb+8+4hi..+3
            v4f b01 = sx4[q];
            v4f b23 = sx4[q + 2];
            acc0 = __builtin_amdgcn_wmma_f32_16x16x4_f32(
                false, a01.lo, false, b01.lo, (short)0, acc0, false, false);
            acc1 = __builtin_amdgcn_wmma_f32_16x16x4_f32(
                false, a01.hi, false, b01.hi, (short)0, acc1, false, false);
            acc2 = __builtin_amdgcn_wmma_f32_16x16x4_f32(
                false, a23.lo, false, b23.lo, (short)0, acc2, false, false);
            acc3 = __builtin_amdgcn_wmma_f32_16x16x4_f32(
                false, a23.hi, false, b23.hi, (short)0, acc3, false, false);
        }

        v8f acc;
#pragma unroll
        for (int j = 0; j < 8; ++j) {
            acc[j] = (acc0[j] + acc1[j]) + (acc2[j] + acc3[j]);
        }

        if (row == 0) {  // lanes 0 and 16 hold column N=0
            const int mofs = hi * 8;
#pragma unroll
            for (int j = 0; j < 8; ++j) {
                const int m = rbase + mofs + j;
                if (m < M) {
                    float v = acc[j] + bias[m];
                    if (relu_flag) v = fmaxf(v, 0.0f);
                    y[m] = v;
                }
            }
        }
    }
}

// ---------------------------------------------------------------------------
// GRU gates (PyTorch order r, z, n): h_new = (1-z)*n + z*h0
// ---------------------------------------------------------------------------
__global__ void gru_gates_kernel(const float* __restrict__ gx,
                                 const float* __restrict__ gh,
                                 const float* __restrict__ h0,
                                 float* __restrict__ hnew_ws,
                                 float* __restrict__ out_h) {
    const int c = blockIdx.x * blockDim.x + threadIdx.x;
    if (c >= H_DIM) return;
    const float r = 1.0f / (1.0f + expf(-(gx[c] + gh[c])));
    const float z = 1.0f / (1.0f + expf(-(gx[H_DIM + c] + gh[H_DIM + c])));
    const float n = tanhf(gx[2 * H_DIM + c] + r * gh[2 * H_DIM + c]);
    const float h = (1.0f - z) * n + z * h0[c];
    hnew_ws[c] = h;
    out_h[c]   = h;
}

// ---------------------------------------------------------------------------
// Single-block log-softmax reduction over V: red[0] = max, red[1] = log(sum exp)
// ---------------------------------------------------------------------------
__global__ void softmax_reduce_kernel(const float* __restrict__ logits,
                                      float* __restrict__ red, int V) {
    __shared__ float sm[1024];
    const int tid = threadIdx.x;
    const int nt  = blockDim.x;

    float m = -INFINITY;
    for (int i = tid; i < V; i += nt) m = fmaxf(m, logits[i]);
    sm[tid] = m;
    __syncthreads();
    for (int s = nt >> 1; s > 0; s >>= 1) {
        if (tid < s) sm[tid] = fmaxf(sm[tid], sm[tid + s]);
        __syncthreads();
    }
    const float maxv = sm[0];
    __syncthreads();

    float sum = 0.0f;
    for (int i = tid; i < V; i += nt) sum += expf(logits[i] - maxv);
    sm[tid] = sum;
    __syncthreads();
    for (int s = nt >> 1; s > 0; s >>= 1) {
        if (tid < s) sm[tid] += sm[tid + s];
        __syncthreads();
    }
    if (tid == 0) {
        red[0] = maxv;
        red[1] = logf(sm[0]);
    }
}

__global__ void softmax_write_kernel(const float* __restrict__ logits,
                                     const float* __restrict__ red,
                                     float* __restrict__ out, int V) {
    const int i = blockIdx.x * blockDim.x + threadIdx.x;
    if (i < V) out[i] = logits[i] - red[0] - red[1];
}

// ---------------------------------------------------------------------------
// Host-side orchestration
// ---------------------------------------------------------------------------
extern "C" void kernel_launch(void* const* d_in, const int* in_sizes, int n_in,
                              void* d_out, int out_size, void* d_ws, size_t ws_size,
                              hipStream_t stream) {
    const int*   tok    = (const int*)  d_in[0];
    const float* hidden = (const float*)d_in[1];   // (1,1,H)
    const float* enc    = (const float*)d_in[2];   // (S,H)
    const float* emb    = (const float*)d_in[3];   // (V,H)
    // d_in[4] attn_w, d_in[5] attn_b: unused (softmax over singleton == 1)
    const float* comb_w = (const float*)d_in[6];   // (H,2H)
    const float* comb_b = (const float*)d_in[7];
    const float* w_ih   = (const float*)d_in[8];   // (3H,H)
    const float* w_hh   = (const float*)d_in[9];   // (3H,H)
    const float* b_ih   = (const float*)d_in[10];
    const float* b_hh   = (const float*)d_in[11];
    const float* out_w  = (const float*)d_in[12];  // (V,H)
    const float* out_b  = (const float*)d_in[13];

    float* out        = (float*)d_out;
    float* out_logp   = out;                       // [0, V)
    float* out_hnew   = out + V_DIM;               // [V, V+H)
    float* out_attnw  = out + V_DIM + H_DIM;       // [V+H, V+H+S)

    // Workspace layout (floats, all offsets 16B-aligned)
    float* ws       = (float*)d_ws;
    float* partials = ws;                          // 32*H = 65536
    float* c2h      = partials + 32 * H_DIM;       // 2H   = 4096
    float* xbuf     = c2h + 2 * H_DIM;             // H
    float* gx       = xbuf + H_DIM;                // 3H
    float* gh       = gx + 3 * H_DIM;              // 3H
    float* hnew     = gh + 3 * H_DIM;              // H
    float* logits   = hnew + H_DIM;                // V
    float* red      = logits + V_DIM + 15;         // 2 scalars

    // 1) encoder column-sum partials (16 MB streamed, b128 coalesced)
    colsum_partial_kernel<<<dim3(H_DIM / 4 / 256, 32), 256, 0, stream>>>(enc, partials);

    // 2) combine + embedding gather + attn_weights(=1) output
    prep_combine_kernel<<<H_DIM / 256, 256, 0, stream>>>(emb, tok, partials, c2h, out_attnw);

    // 3) x = relu(comb_w @ c2h + comb_b)     (M=2048, K=4096 -> 128 groups)
    gemv_wmma_kernel<<<16, 256, 2 * H_DIM * sizeof(float), stream>>>(
        comb_w, c2h, comb_b, xbuf, H_DIM, 2 * H_DIM, 1);

    // 4) gx = w_ih @ x + b_ih                (M=6144, K=2048 -> 384 groups)
    gemv_wmma_kernel<<<48, 256, H_DIM * sizeof(float), stream>>>(
        w_ih, xbuf, b_ih, gx, 3 * H_DIM, H_DIM, 0);

    // 5) gh = w_hh @ h0 + b_hh
    gemv_wmma_kernel<<<48, 256, H_DIM * sizeof(float), stream>>>(
        w_hh, hidden, b_hh, gh, 3 * H_DIM, H_DIM, 0);

    // 6) GRU gate math -> h_new (workspace + output)
    gru_gates_kernel<<<H_DIM / 256, 256, 0, stream>>>(gx, gh, hidden, hnew, out_hnew);

    // 7) logits = out_w @ h_new + out_b      (M=50257, K=2048 -> 3142 groups, 412 MB)
    gemv_wmma_kernel<<<393, 256, H_DIM * sizeof(float), stream>>>(
        out_w, hnew, out_b, logits, V_DIM, H_DIM, 0);

    // 8) log-softmax reduction + write
    softmax_reduce_kernel<<<1, 1024, 0, stream>>>(logits, red, V_DIM);
    softmax_write_kernel<<<(V_DIM + 255) / 256, 256, 0, stream>>>(logits, red,
                                                                  out_logp, V_DIM);
}